// GeneratorHierarchical0_82480551952938
// MI455X (gfx1250) — compile-verified
//
#include <hip/hip_runtime.h>
#include <hip/hip_bf16.h>

typedef float v2f __attribute__((ext_vector_type(2)));
typedef float v8f __attribute__((ext_vector_type(8)));

#define NODES_OUT 67615

struct Params {
  const float* z;
  const int *svec, *tvec, *cvec;
  const float *study, *task, *contrast;
  const float *fc0w, *fc0b, *fc1w, *fc1b, *fc2w, *fc2b, *fc3w, *fc3b, *fc4w, *fc4b;
  const float *w0, *b0, *w1, *b1, *w2, *b2, *w3, *b3, *w4, *b4;
  const float *g0, *be0, *g1, *be1, *g2, *be2, *g3, *be3;
};

// One 16x16 D-tile pair (M=32 rows) per N-tile, K stepped by 4 using f32 WMMA.
// A-layout (16x4 f32): lane l in [0,16): M=l; v0=K0/K2 (half), v1=K1/K3.
// B-layout (4x16 f32): mirror of A with N=lane&15.
// C/D layout: VGPR i holds M=i+8*half, N=lane&15.
template <int CIN, int O>
__device__ __forceinline__ void level_gemm(float (*X)[152], float (*H)[68],
                                           const float* __restrict__ W,
                                           const float* __restrict__ Bias,
                                           int lane) {
  const int l = lane & 15;
  const int half = lane >> 4;
  constexpr int NT = (O + 15) / 16;
#pragma unroll
  for (int nt = 0; nt < NT; ++nt) {
    const int o = nt * 16 + l;
    const bool ok = (o < O);
#pragma unroll
    for (int mt = 0; mt < 2; ++mt) {
      v8f acc = {0.f, 0.f, 0.f, 0.f, 0.f, 0.f, 0.f, 0.f};
      for (int k = 0; k < CIN; k += 4) {
        v2f a, b;
        a[0] = X[mt * 16 + l][k + 2 * half];
        a[1] = X[mt * 16 + l][k + 1 + 2 * half];
        b[0] = ok ? W[o * CIN + k + 2 * half] : 0.f;
        b[1] = ok ? W[o * CIN + k + 1 + 2 * half] : 0.f;
        acc = __builtin_amdgcn_wmma_f32_16x16x4_f32(false, a, false, b,
                                                    (short)0, acc, false, false);
      }
      if (ok) {
        const float bias = Bias[o];
#pragma unroll
        for (int i = 0; i < 8; ++i)
          H[mt * 16 + i + 8 * half][o] = acc[i] + bias;
      }
    }
  }
}

// Leaky ReLU then batchnorm over the batch axis (node axis is constant, so
// stats over (batch, nodes) == stats over batch). One lane owns one channel.
template <int O>
__device__ __forceinline__ void bn_leaky(float (*H)[68], const float* __restrict__ g,
                                         const float* __restrict__ be, int lane) {
  for (int o = lane; o < O; o += 32) {
    float s = 0.f, s2 = 0.f;
    for (int n = 0; n < 32; ++n) {
      float v = H[n][o];
      v = v >= 0.f ? v : 0.2f * v;
      H[n][o] = v;
      s += v;
      s2 += v * v;
    }
    const float mean = s * (1.f / 32.f);
    const float var = s2 * (1.f / 32.f) - mean * mean;
    const float sc = g[o] * rsqrtf(var + 1e-5f);
    const float sh = be[o] - mean * sc;
    for (int n = 0; n < 32; ++n) H[n][o] = H[n][o] * sc + sh;
  }
}

__device__ __forceinline__ void fc16(float (*emb)[48], float (*dst)[16],
                                     const float* __restrict__ w,
                                     const float* __restrict__ b, int K, int lane) {
  for (int c = 0; c < 16; ++c) {
    float acc = b[c];
    for (int k = 0; k < K; ++k) acc += emb[lane][k] * w[k * 16 + c];
    dst[lane][c] = acc;
  }
}

__global__ __launch_bounds__(32) void mlp_kernel(Params p, float* __restrict__ val) {
  __shared__ float emb[32][48];
  __shared__ float cont[5][32][16];
  __shared__ float X[32][152];
  __shared__ float H[32][68];
  const int lane = threadIdx.x;

  // Gather embeddings: lane n = batch row n.
  {
    const int s = p.svec[lane], t = p.tvec[lane], c = p.cvec[lane];
    for (int k = 0; k < 16; ++k) {
      emb[lane][k] = p.study[s * 16 + k];
      emb[lane][16 + k] = p.task[t * 16 + k];
      emb[lane][32 + k] = p.contrast[c * 16 + k];
    }
  }
  // Per-level content vectors (tiny FCs), all per-lane (no cross-lane reads).
  fc16(emb, cont[0], p.fc0w, p.fc0b, 16, lane);
  fc16(emb, cont[1], p.fc1w, p.fc1b, 32, lane);
  fc16(emb, cont[2], p.fc2w, p.fc2b, 48, lane);
  fc16(emb, cont[3], p.fc3w, p.fc3b, 48, lane);
  fc16(emb, cont[4], p.fc4w, p.fc4b, 48, lane);

  // Level-0 input: [z | content0]  (node axis is provably constant -> dropped)
  for (int c = 0; c < 128; ++c) X[lane][c] = p.z[lane * 128 + c];
  for (int c = 0; c < 16; ++c) X[lane][128 + c] = cont[0][lane][c];
  __syncthreads();

  level_gemm<144, 64>(X, H, p.w0, p.b0, lane);
  __syncthreads();
  bn_leaky<64>(H, p.g0, p.be0, lane);
  __syncthreads();
  for (int c = 0; c < 64; ++c) X[lane][c] = H[lane][c];
  for (int c = 0; c < 16; ++c) X[lane][64 + c] = cont[1][lane][c];
  __syncthreads();

  level_gemm<80, 32>(X, H, p.w1, p.b1, lane);
  __syncthreads();
  bn_leaky<32>(H, p.g1, p.be1, lane);
  __syncthreads();
  for (int c = 0; c < 32; ++c) X[lane][c] = H[lane][c];
  for (int c = 0; c < 16; ++c) X[lane][32 + c] = cont[2][lane][c];
  __syncthreads();

  level_gemm<48, 16>(X, H, p.w2, p.b2, lane);
  __syncthreads();
  bn_leaky<16>(H, p.g2, p.be2, lane);
  __syncthreads();
  for (int c = 0; c < 16; ++c) X[lane][c] = H[lane][c];
  for (int c = 0; c < 16; ++c) X[lane][16 + c] = cont[3][lane][c];
  __syncthreads();

  level_gemm<32, 8>(X, H, p.w3, p.b3, lane);
  __syncthreads();
  bn_leaky<8>(H, p.g3, p.be3, lane);
  __syncthreads();
  for (int c = 0; c < 8; ++c) X[lane][c] = H[lane][c];
  for (int c = 0; c < 16; ++c) X[lane][8 + c] = cont[4][lane][c];
  __syncthreads();

  level_gemm<24, 1>(X, H, p.w4, p.b4, lane);
  __syncthreads();
  val[lane] = tanhf(H[lane][0]);
}

// Store-bound broadcast: out[n, j] = val[n] for all 67615 nodes.
__global__ __launch_bounds__(256) void bcast_kernel(const float* __restrict__ val,
                                                    float* __restrict__ out) {
  const int n = blockIdx.y;
  const float v = val[n];
  const int j = blockIdx.x * blockDim.x + threadIdx.x;
  if (j < NODES_OUT) out[(size_t)n * NODES_OUT + j] = v;
}

extern "C" void kernel_launch(void* const* d_in, const int* in_sizes, int n_in,
                              void* d_out, int out_size, void* d_ws, size_t ws_size,
                              hipStream_t stream) {
  (void)in_sizes; (void)n_in; (void)out_size; (void)ws_size;
  Params p;
  p.z = (const float*)d_in[0];
  p.svec = (const int*)d_in[1];
  p.tvec = (const int*)d_in[2];
  p.cvec = (const int*)d_in[3];
  p.study = (const float*)d_in[4];
  p.task = (const float*)d_in[5];
  p.contrast = (const float*)d_in[6];
  p.fc0w = (const float*)d_in[7];  p.fc0b = (const float*)d_in[8];
  p.fc1w = (const float*)d_in[9];  p.fc1b = (const float*)d_in[10];
  p.fc2w = (const float*)d_in[11]; p.fc2b = (const float*)d_in[12];
  p.fc3w = (const float*)d_in[13]; p.fc3b = (const float*)d_in[14];
  p.fc4w = (const float*)d_in[15]; p.fc4b = (const float*)d_in[16];
  p.w0 = (const float*)d_in[17]; p.b0 = (const float*)d_in[18];
  p.w1 = (const float*)d_in[19]; p.b1 = (const float*)d_in[20];
  p.w2 = (const float*)d_in[21]; p.b2 = (const float*)d_in[22];
  p.w3 = (const float*)d_in[23]; p.b3 = (const float*)d_in[24];
  p.w4 = (const float*)d_in[25]; p.b4 = (const float*)d_in[26];
  p.g0 = (const float*)d_in[27]; p.be0 = (const float*)d_in[28];
  p.g1 = (const float*)d_in[29]; p.be1 = (const float*)d_in[30];
  p.g2 = (const float*)d_in[31]; p.be2 = (const float*)d_in[32];
  p.g3 = (const float*)d_in[33]; p.be3 = (const float*)d_in[34];
  // d_in[35..39] = p0..p4: gathers of a node-constant tensor are identity -> unused.

  float* val = (float*)d_ws;  // 32 floats of scratch
  mlp_kernel<<<dim3(1), dim3(32), 0, stream>>>(p, val);
  bcast_kernel<<<dim3((NODES_OUT + 255) / 256, 32), dim3(256), 0, stream>>>(
      val, (float*)d_out);
}